// MGCNLayer_17532056502541
// MI455X (gfx1250) — compile-verified
//
#include <hip/hip_runtime.h>
#include <hip/hip_bf16.h>

typedef float v2f __attribute__((ext_vector_type(2)));
typedef float v8f __attribute__((ext_vector_type(8)));

#define BN_EPS_F 1e-5f

// ---------------- small utility kernels ----------------

__global__ void k_fill(float* __restrict__ p, float v, int n) {
  int i = blockIdx.x * blockDim.x + threadIdx.x;
  if (i < n) p[i] = v;
}

__global__ void k_deg_accum(float* __restrict__ deg, const int* __restrict__ col,
                            const float* __restrict__ w, int E) {
  int e = blockIdx.x * blockDim.x + threadIdx.x;
  if (e < E)
    __hip_atomic_fetch_add(&deg[col[e]], w[e], __ATOMIC_RELAXED, __HIP_MEMORY_SCOPE_AGENT);
}

__global__ void k_rsqrt_ip(float* __restrict__ p, int n) {
  int i = blockIdx.x * blockDim.x + threadIdx.x;
  if (i < n) p[i] = rsqrtf(p[i]);   // deg > 0 always (self loop +1)
}

__global__ void k_norm(float* __restrict__ nrm, const int* __restrict__ row,
                       const int* __restrict__ col, const float* __restrict__ w,
                       const float* __restrict__ dinv, int E) {
  int e = blockIdx.x * blockDim.x + threadIdx.x;
  if (e < E) nrm[e] = dinv[row[e]] * w[e] * dinv[col[e]];
}

// ---------------- fp32 WMMA GEMM: out[N,128] = X[N,128] @ W[128,128] (+bias) ----
// One 16x16 output tile per wave via V_WMMA_F32_16X16X4_F32, 32 wmma per tile.
// grid.x = N/16 m-tiles; block = 256 threads = 8 waves = 8 n-tiles.
// EXEC is all-ones (no divergence before WMMA), as required by the ISA.
__global__ void k_gemm_xw(const float* __restrict__ X, const float* __restrict__ W,
                          const float* __restrict__ bias, float* __restrict__ out) {
  const int wave = threadIdx.x >> 5;
  const int lane = threadIdx.x & 31;
  const int m0 = blockIdx.x << 4;
  const int n0 = wave << 4;
  const int mr = lane & 15;
  const int kh = (lane >> 4) << 1;                 // 0 (lanes 0-15) or 2 (lanes 16-31)
  const float* xrow = X + (m0 + mr) * 128;
  const float* wcol = W + n0 + (lane & 15);
  v8f c = {};
#pragma unroll
  for (int k = 0; k < 128; k += 4) {
    v2f a, b;
    // A 16x4 (ISA 32-bit A layout): VGPR0 = K = kh, VGPR1 = K = kh+1
    a.x = xrow[k + kh];
    a.y = xrow[k + kh + 1];
    // B 4x16: lanes hold N; VGPR0 = row K = kh, VGPR1 = row K = kh+1
    b.x = wcol[(k + kh) * 128];
    b.y = wcol[(k + kh + 1) * 128];
    c = __builtin_amdgcn_wmma_f32_16x16x4_f32(false, a, false, b, (short)0, c, false, false);
  }
  const int nc = n0 + (lane & 15);
  const float bv = bias ? bias[nc] : 0.0f;
  const int rb = m0 + ((lane >> 4) << 3);          // C/D: VGPR i -> rows i / i+8
#pragma unroll
  for (int i = 0; i < 8; ++i)
    out[(rb + i) * 128 + nc] = c[i] + bv;
}

// ---------------- edge scatter: one wave per edge, lane handles 4 channels ----
__global__ void k_scatter(float* __restrict__ acc, const float* __restrict__ xl,
                          const int* __restrict__ row, const int* __restrict__ col,
                          const float* __restrict__ nrm, int E) {
  const int t = blockIdx.x * blockDim.x + threadIdx.x;
  const int e = t >> 5;
  const int lane = t & 31;
  if (e >= E) return;
  const int rs = row[e];
  const int cd = col[e];
  const float nv = nrm[e];
  const float4 v = ((const float4*)(xl + rs * 128))[lane];   // L2-resident gather
  float* d = acc + cd * 128 + (lane << 2);
  __hip_atomic_fetch_add(d + 0, v.x * nv, __ATOMIC_RELAXED, __HIP_MEMORY_SCOPE_AGENT);
  __hip_atomic_fetch_add(d + 1, v.y * nv, __ATOMIC_RELAXED, __HIP_MEMORY_SCOPE_AGENT);
  __hip_atomic_fetch_add(d + 2, v.z * nv, __ATOMIC_RELAXED, __HIP_MEMORY_SCOPE_AGENT);
  __hip_atomic_fetch_add(d + 3, v.w * nv, __ATOMIC_RELAXED, __HIP_MEMORY_SCOPE_AGENT);
}

// ---------------- self loop + conv bias: acc += xl * dinv^2 + b -----------------
__global__ void k_selfloop(float* __restrict__ acc, const float* __restrict__ xl,
                           const float* __restrict__ dinv, const float* __restrict__ bias,
                           int N) {
  const int q = blockIdx.x * blockDim.x + threadIdx.x;   // over N*32 float4 groups
  if (q >= N * 32) return;
  const int n = q >> 5;
  const int db = (q & 31) << 2;
  const float s = dinv[n] * dinv[n];
  const float4 v = ((const float4*)xl)[q];
  const float4 b = *(const float4*)(bias + db);
  float4 o = ((const float4*)acc)[q];
  o.x += v.x * s + b.x;
  o.y += v.y * s + b.y;
  o.z += v.z * s + b.z;
  o.w += v.w * s + b.w;
  ((float4*)acc)[q] = o;
}

// ---------------- BN stats: per-channel sum / sumsq ----------------------------
__global__ void k_bnstats(const float* __restrict__ acc, float* __restrict__ stats,
                          int total) {
  __shared__ float ssum[128];
  __shared__ float ssq[128];
  if (threadIdx.x < 128) ssum[threadIdx.x] = 0.f;
  else                   ssq[threadIdx.x - 128] = 0.f;
  __syncthreads();
  const int stride = gridDim.x * blockDim.x;             // multiple of 128
  int idx = blockIdx.x * blockDim.x + threadIdx.x;
  const int d = idx & 127;                               // fixed channel per thread
  float s = 0.f, q = 0.f;
  for (; idx < total; idx += stride) {
    const float v = acc[idx];
    s += v;
    q += v * v;
  }
  atomicAdd(&ssum[d], s);
  atomicAdd(&ssq[d], q);
  __syncthreads();
  if (threadIdx.x < 128)
    __hip_atomic_fetch_add(&stats[threadIdx.x], ssum[threadIdx.x],
                           __ATOMIC_RELAXED, __HIP_MEMORY_SCOPE_AGENT);
  else
    __hip_atomic_fetch_add(&stats[threadIdx.x], ssq[threadIdx.x - 128],
                           __ATOMIC_RELAXED, __HIP_MEMORY_SCOPE_AGENT);
}

__global__ void k_bnparams(const float* __restrict__ stats, float* __restrict__ mr,
                           float invN) {
  const int d = threadIdx.x;                             // 128 threads
  const float mean = stats[d] * invN;
  const float var = stats[128 + d] * invN - mean * mean; // biased var (jnp.var)
  mr[d] = mean;
  mr[128 + d] = rsqrtf(var + BN_EPS_F);
}

__global__ void k_bnapply(float* __restrict__ out, const float* __restrict__ acc,
                          const float* __restrict__ mr, const float* __restrict__ gamma,
                          const float* __restrict__ beta, int N) {
  const int q = blockIdx.x * blockDim.x + threadIdx.x;   // N*32 float4 groups
  if (q >= N * 32) return;
  const int db = (q & 31) << 2;
  const float4 v = ((const float4*)acc)[q];
  const float4 m = *(const float4*)(mr + db);
  const float4 r = *(const float4*)(mr + 128 + db);
  const float4 g = *(const float4*)(gamma + db);
  const float4 b = *(const float4*)(beta + db);
  float4 o;
  o.x = fmaxf((v.x - m.x) * r.x * g.x + b.x, 0.f);
  o.y = fmaxf((v.y - m.y) * r.y * g.y + b.y, 0.f);
  o.z = fmaxf((v.z - m.z) * r.z * g.z + b.z, 0.f);
  o.w = fmaxf((v.w - m.w) * r.w * g.w + b.w, 0.f);
  ((float4*)out)[q] = o;
}

// ---------------- launcher -----------------------------------------------------

extern "C" void kernel_launch(void* const* d_in, const int* in_sizes, int n_in,
                              void* d_out, int out_size, void* d_ws, size_t ws_size,
                              hipStream_t stream) {
  const float* x     = (const float*)d_in[0];
  const int*   ei    = (const int*)  d_in[1];   // [R,2,E]
  const float* es    = (const float*)d_in[2];   // [R,E]
  const float* Wc    = (const float*)d_in[3];   // [R,128,128]
  const float* bc    = (const float*)d_in[4];   // [R,128]
  const float* Ws    = (const float*)d_in[5];   // [128,128]
  const float* bs    = (const float*)d_in[6];   // [128]
  const float* gamma = (const float*)d_in[7];
  const float* beta  = (const float*)d_in[8];
  float* out = (float*)d_out;

  const int R = 4;
  const int N = in_sizes[0] / 128;
  const int E = in_sizes[2] / R;

  // workspace layout (floats): acc[N*128] | xl[N*128] | deg/dinv[N] | nrm[E] | stats[256] | mr[256]
  float* ws    = (float*)d_ws;
  float* acc   = ws;
  float* xl    = acc + (size_t)N * 128;
  float* deg   = xl + (size_t)N * 128;
  float* nrm   = deg + N;
  float* stats = nrm + E;
  float* mr    = stats + 256;

  const int TB = 256;
  const int mtiles = N / 16;                       // 50000/16 = 3125

  // acc = x @ W_skip + b_skip
  k_gemm_xw<<<mtiles, TB, 0, stream>>>(x, Ws, bs, acc);

  for (int r = 0; r < R; ++r) {
    const int*   row = ei + (size_t)r * 2 * E;     // source
    const int*   col = row + E;                    // target / aggregation index
    const float* w   = es + (size_t)r * E;

    k_fill<<<(N + TB - 1) / TB, TB, 0, stream>>>(deg, 1.0f, N);          // self loop
    k_deg_accum<<<(E + TB - 1) / TB, TB, 0, stream>>>(deg, col, w, E);
    k_rsqrt_ip<<<(N + TB - 1) / TB, TB, 0, stream>>>(deg, N);            // deg -> dinv
    k_gemm_xw<<<mtiles, TB, 0, stream>>>(x, Wc + (size_t)r * 128 * 128, nullptr, xl);
    k_norm<<<(E + TB - 1) / TB, TB, 0, stream>>>(nrm, row, col, w, deg, E);

    const long long sthreads = (long long)E * 32;
    k_scatter<<<(unsigned)((sthreads + TB - 1) / TB), TB, 0, stream>>>(acc, xl, row, col, nrm, E);
    k_selfloop<<<(N * 32 + TB - 1) / TB, TB, 0, stream>>>(acc, xl, deg,
                                                          bc + (size_t)r * 128, N);
  }

  k_fill<<<1, TB, 0, stream>>>(stats, 0.0f, 256);
  k_bnstats<<<1024, TB, 0, stream>>>(acc, stats, N * 128);
  k_bnparams<<<1, 128, 0, stream>>>(stats, mr, 1.0f / (float)N);
  k_bnapply<<<(N * 32 + TB - 1) / TB, TB, 0, stream>>>(out, acc, mr, gamma, beta, N);
}